// GMML_65755949302327
// MI455X (gfx1250) — compile-verified
//
#include <hip/hip_runtime.h>
#include <math.h>

// Problem sizes (fixed by the reference)
#define Bn 1024
#define Sn 256
#define Dn 256

typedef __attribute__((ext_vector_type(2))) float v2f;
typedef __attribute__((ext_vector_type(8))) float v8f;

// D = A(16x4) * B(4x16) + C, FP32 WMMA (CDNA5, wave32).
__device__ __forceinline__ v8f wmma_f32_4(v2f a, v2f b, v8f c) {
    return __builtin_amdgcn_wmma_f32_16x16x4_f32(
        false, a, false, b, (short)0, c, false, false);
}

__device__ __forceinline__ v2f ld2(const float* p) {
    return *reinterpret_cast<const v2f*>(p);   // 8B-aligned by construction
}

__device__ __forceinline__ v8f vzero8() {
    v8f z;
    #pragma unroll
    for (int q = 0; q < 8; ++q) z[q] = 0.f;
    return z;
}

// ---------------------------------------------------------------------------
// Kernel 1: Sigma[s] = sp[s]^T sp[s], 32x32 output tile per wave (4 C frags,
// each A/B fragment reused twice -> 2 loads per WMMA instead of 4).
// FP32 frag layout: lanes 0-15 hold K = k0,k0+1 ; lanes 16-31 hold K = k0+2,k0+3
// ---------------------------------------------------------------------------
__global__ __launch_bounds__(32) void gram_kernel(const float* __restrict__ sp,
                                                  float* __restrict__ Sig) {
    const int s  = blockIdx.z;
    const int d0 = blockIdx.x * 32;
    const int e0 = blockIdx.y * 32;
    const float* S0 = sp + (size_t)s * Dn * Dn;
    const int lane = threadIdx.x;
    const int half = lane >> 4;
    const int r    = lane & 15;

    v8f c00 = vzero8(), c01 = vzero8(), c10 = vzero8(), c11 = vzero8();

    for (int k0 = 0; k0 < Dn; k0 += 4) {
        const float* row0 = S0 + (size_t)(k0 + 2 * half) * Dn;
        const float* row1 = row0 + Dn;
        if (k0 + 4 < Dn)                       // prefetch next k-panel (HBM stream)
            __builtin_prefetch(row0 + 4 * (size_t)Dn + d0 + r, 0, 1);
        v2f a0, a1, b0, b1;
        a0.x = row0[d0 + r];       a0.y = row1[d0 + r];
        a1.x = row0[d0 + 16 + r];  a1.y = row1[d0 + 16 + r];
        b0.x = row0[e0 + r];       b0.y = row1[e0 + r];
        b1.x = row0[e0 + 16 + r];  b1.y = row1[e0 + 16 + r];
        c00 = wmma_f32_4(a0, b0, c00);
        c01 = wmma_f32_4(a0, b1, c01);
        c10 = wmma_f32_4(a1, b0, c10);
        c11 = wmma_f32_4(a1, b1, c11);
    }

    float* out = Sig + (size_t)s * Dn * Dn;
    #pragma unroll
    for (int q = 0; q < 8; ++q) {
        const int m0 = d0 + q + 8 * half;
        out[(size_t)m0        * Dn + e0 + r]      = c00[q];
        out[(size_t)m0        * Dn + e0 + 16 + r] = c01[q];
        out[(size_t)(m0 + 16) * Dn + e0 + r]      = c10[q];
        out[(size_t)(m0 + 16) * Dn + e0 + 16 + r] = c11[q];
    }
}

// ---------------------------------------------------------------------------
// Kernel 2: ridge = 0.01 * ||sp_s||_F^2 / D (== 0.01*trace(Gram)/D), add to diag
// ---------------------------------------------------------------------------
__global__ __launch_bounds__(256) void ridge_kernel(const float* __restrict__ sp,
                                                    float* __restrict__ Sig) {
    const int s = blockIdx.x;
    const int t = threadIdx.x;
    const float* S0 = sp + (size_t)s * Dn * Dn;

    float acc = 0.f;
    for (int i = t; i < Dn * Dn; i += 256) { float v = S0[i]; acc += v * v; }

    __shared__ float red[256];
    red[t] = acc;
    __syncthreads();
    for (int off = 128; off > 0; off >>= 1) {
        if (t < off) red[t] += red[t + off];
        __syncthreads();
    }
    const float ridge = 0.01f * red[0] / (float)Dn;
    Sig[(size_t)s * Dn * Dn + (size_t)t * Dn + t] += ridge;
}

// ---------------------------------------------------------------------------
// Kernel 3: in-place Cholesky (lower) per state + logdet. Upper triangle is
// never read afterwards, so it is left as-is.
// ---------------------------------------------------------------------------
__global__ __launch_bounds__(256) void chol_kernel(float* __restrict__ Sig,
                                                   float* __restrict__ logdet) {
    const int s = blockIdx.x;
    const int i = threadIdx.x;
    float* M = Sig + (size_t)s * Dn * Dn;

    __shared__ float col[Dn];
    __shared__ float sdiag;
    float ld = 0.f;

    for (int j = 0; j < Dn; ++j) {
        if (i == 0) {
            float d = sqrtf(M[(size_t)j * Dn + j]);
            M[(size_t)j * Dn + j] = d;
            sdiag = d;
            ld += 2.f * logf(d);
        }
        __syncthreads();
        const float d = sdiag;
        if (i > j) {
            float l = M[(size_t)i * Dn + j] / d;
            M[(size_t)i * Dn + j] = l;
            col[i] = l;
        }
        __syncthreads();
        if (i > j) {
            const float li = col[i];
            for (int k = j + 1; k <= i; ++k)
                M[(size_t)i * Dn + k] -= li * col[k];
        }
        __syncthreads();
    }
    if (i == 0) logdet[s] = ld;
}

// ---------------------------------------------------------------------------
// Kernel 4: in-place blocked triangular inverse W = L^{-1} (one wave / state).
// Phase A: invert all 16 diagonal 16x16 blocks (scalar fwd-subst, zero upper).
// Phase B: column sweep, W_ij = -W_ii * (sum_{k=j}^{i-1} L_ik W_kj), both
//          GEMMs on v_wmma_f32_16x16x4_f32; A-fragments via b64 loads,
//          intermediate re-staged through LDS.
// ---------------------------------------------------------------------------
__global__ __launch_bounds__(32) void trtri_kernel(float* __restrict__ Sig) {
    const int s = blockIdx.x;
    float* M = Sig + (size_t)s * Dn * Dn;
    const int lane = threadIdx.x;
    const int half = lane >> 4;
    const int r    = lane & 15;

    __shared__ float Lb[256];
    __shared__ float Wb[256];
    __shared__ float T[256];

    // Phase A: diagonal block inversion (in place)
    for (int J = 0; J < 16; ++J) {
        float* Bl = M + (size_t)(16 * J) * Dn + 16 * J;
        for (int e = lane; e < 256; e += 32)
            Lb[e] = Bl[(size_t)(e >> 4) * Dn + (e & 15)];
        __syncthreads();
        if (lane < 16) {
            const int c = lane;                 // this lane solves column c
            for (int ii = 0; ii < 16; ++ii) {
                float w;
                if (ii < c) {
                    w = 0.f;                    // clean zero upper-in-tile
                } else {
                    float sacc = (ii == c) ? 1.f : 0.f;
                    for (int k = c; k < ii; ++k)
                        sacc -= Lb[ii * 16 + k] * Wb[k * 16 + c];
                    w = sacc / Lb[ii * 16 + ii];
                }
                Wb[ii * 16 + c] = w;
            }
        }
        __syncthreads();
        for (int e = lane; e < 256; e += 32)
            Bl[(size_t)(e >> 4) * Dn + (e & 15)] = Wb[e];
        __syncthreads();
    }

    // Phase B: column sweep over off-diagonal blocks
    for (int j = 0; j < 16; ++j) {
        for (int i = j + 1; i < 16; ++i) {
            v8f acc = vzero8();
            for (int k = j; k < i; ++k) {
                #pragma unroll
                for (int kk0 = 0; kk0 < 16; kk0 += 4) {
                    const int kr = kk0 + 2 * half;
                    v2f a = ld2(&M[(size_t)(16 * i + r) * Dn + 16 * k + kr]);
                    v2f b;
                    b.x = M[(size_t)(16 * k + kr)     * Dn + 16 * j + r];
                    b.y = M[(size_t)(16 * k + kr + 1) * Dn + 16 * j + r];
                    acc = wmma_f32_4(a, b, acc);
                }
            }
            // Re-stage acc (C-frag layout) through LDS as a B-fragment source
            #pragma unroll
            for (int q = 0; q < 8; ++q)
                T[(q + 8 * half) * 16 + r] = acc[q];
            __syncthreads();

            // res = W_ii @ acc ; store W_ij = -res
            v8f res = vzero8();
            #pragma unroll
            for (int kk0 = 0; kk0 < 16; kk0 += 4) {
                const int kr = kk0 + 2 * half;
                v2f a = ld2(&M[(size_t)(16 * i + r) * Dn + 16 * i + kr]);
                v2f b;
                b.x = T[kr * 16 + r];
                b.y = T[(kr + 1) * 16 + r];
                res = wmma_f32_4(a, b, res);
            }
            #pragma unroll
            for (int q = 0; q < 8; ++q)
                M[(size_t)(16 * i + q + 8 * half) * Dn + 16 * j + r] = -res[q];
            __syncthreads();
        }
    }
}

// ---------------------------------------------------------------------------
// Kernel 5: y = W (x_b - mu_s); maha = ||y||^2;
//           logp = -0.5(D log2pi + logdet + maha), stored transposed to (B,S).
// One wave per (state, 32-column strip of B); row tiles processed in pairs
// (32x32 register block, 4 accumulators). Triangular: k-tiles <= row tile,
// enforced with a wave-uniform guard (EXEC stays all-1s for WMMA).
// All fragment loads are contiguous b64; mu fragment shared by both col tiles.
// ---------------------------------------------------------------------------
__global__ __launch_bounds__(32) void logp_kernel(const float* __restrict__ x,
                                                  const float* __restrict__ mu,
                                                  const float* __restrict__ W,
                                                  const float* __restrict__ logdet,
                                                  float* __restrict__ out) {
    const int b0 = blockIdx.x * 32;
    const int s  = blockIdx.y;
    const float* Ws = W  + (size_t)s * Dn * Dn;
    const float* ms = mu + (size_t)s * Dn;
    const int lane = threadIdx.x;
    const int half = lane >> 4;
    const int r    = lane & 15;

    float psum0 = 0.f, psum1 = 0.f;        // col strips [b0, b0+16)

    for (int ii = 0; ii < 8; ++ii) {       // row-tile pair (rows 32*ii .. +31)
        const int i0 = 2 * ii, i1 = i0 + 1;
        v8f y00 = vzero8(), y01 = vzero8();    // rows of tile i0
        v8f y10 = vzero8(), y11 = vzero8();    // rows of tile i1
        for (int k = 0; k <= i1; ++k) {
            const bool doTop = (k <= i0);      // wave-uniform
            #pragma unroll
            for (int kk0 = 0; kk0 < 16; kk0 += 4) {
                const int kr = 16 * k + kk0 + 2 * half;
                const v2f m2  = ld2(&ms[kr]);
                const v2f bb0 = ld2(&x[(size_t)(b0 + r)      * Dn + kr]) - m2;
                const v2f bb1 = ld2(&x[(size_t)(b0 + 16 + r) * Dn + kr]) - m2;
                const v2f a1  = ld2(&Ws[(size_t)(16 * i1 + r) * Dn + kr]);
                y10 = wmma_f32_4(a1, bb0, y10);
                y11 = wmma_f32_4(a1, bb1, y11);
                if (doTop) {
                    const v2f a0 = ld2(&Ws[(size_t)(16 * i0 + r) * Dn + kr]);
                    y00 = wmma_f32_4(a0, bb0, y00);
                    y01 = wmma_f32_4(a0, bb1, y01);
                }
            }
        }
        #pragma unroll
        for (int q = 0; q < 8; ++q) {
            psum0 += y00[q] * y00[q] + y10[q] * y10[q];
            psum1 += y01[q] * y01[q] + y11[q] * y11[q];
        }
    }

    __shared__ float red[64];
    red[lane]      = psum0;
    red[32 + lane] = psum1;
    __syncthreads();
    if (lane < 16) {
        const float LOG2PI = 1.8378770664093453f;
        const float base = (float)Dn * LOG2PI + logdet[s];
        const float maha0 = red[lane]      + red[lane + 16];
        const float maha1 = red[32 + lane] + red[48 + lane];
        out[(size_t)(b0 + lane)      * Sn + s] = -0.5f * (base + maha0);
        out[(size_t)(b0 + 16 + lane) * Sn + s] = -0.5f * (base + maha1);
    }
}

// ---------------------------------------------------------------------------
// Launch: inputs are x (B*D), mu_p (S*1*D), sigma_p (S*1*D*D), all fp32.
// ws layout: Sigma/L/W in-place buffer (S*D*D floats = 64 MiB) | logdet (S).
// ---------------------------------------------------------------------------
extern "C" void kernel_launch(void* const* d_in, const int* in_sizes, int n_in,
                              void* d_out, int out_size, void* d_ws, size_t ws_size,
                              hipStream_t stream) {
    const float* x  = (const float*)d_in[0];
    const float* mu = (const float*)d_in[1];
    const float* sp = (const float*)d_in[2];

    float* Sig    = (float*)d_ws;                        // S*D*D
    float* logdet = Sig + (size_t)Sn * Dn * Dn;          // S
    float* out    = (float*)d_out;                       // B*S

    gram_kernel <<<dim3(Dn / 32, Dn / 32, Sn), 32, 0, stream>>>(sp, Sig);
    ridge_kernel<<<Sn, 256, 0, stream>>>(sp, Sig);
    chol_kernel <<<Sn, 256, 0, stream>>>(Sig, logdet);
    trtri_kernel<<<Sn, 32, 0, stream>>>(Sig);
    logp_kernel <<<dim3(Bn / 32, Sn), 32, 0, stream>>>(x, mu, Sig, logdet, out);
}